// CML2D_21380347199580
// MI455X (gfx1250) — compile-verified
//
#include <hip/hip_runtime.h>
#include <stdint.h>

// CML 2D logistic-map lattice, 15 steps fully fused in LDS.
// grid_new = conv3x3_{K''}(mapped) + BETA*drive,  K'' = (1-B)*E*K + (1-B)(1-E)*delta_center
#define RC      3.9f
#define EPSC    0.3f
#define BETAC   0.15f
#define NSTEPS  15
#define CLAMPC  1e-4f

#define WIMG  512
#define HIMG  512
#define PLANE (WIMG * HIMG)

#define TOUT  128    // output tile (halo redundancy (158/128)^2 = 1.52x)
#define HALO  15     // 1 cell of halo per fused step
#define TIN   158    // TOUT + 2*HALO
#define PITCH 160    // padded pitch; also M guard-ring dims
#define LDSN  (PITCH * PITCH)   // 25600 floats = 102.4KB per array, 307KB total
#define NTHR  512    // 16 wave32 waves -> 4 waves/SIMD on one WGP
#define RPW   10     // rows per wave chunk (16*10 = 160 >= 158)

// CDNA5 async copy: global -> LDS, tracked by ASYNCcnt (no VGPR round-trip).
__device__ __forceinline__ void async_ld_b32(uint32_t lds_off, const float* g) {
  asm volatile("global_load_async_to_lds_b32 %0, %1, off"
               :: "v"(lds_off), "v"(g)
               : "memory");
}

__device__ __forceinline__ void wait_async_0() {
#if __has_builtin(__builtin_amdgcn_s_wait_asynccnt)
  __builtin_amdgcn_s_wait_asynccnt(0);
#else
  asm volatile("s_wait_asynccnt 0" ::: "memory");
#endif
}

__global__ __launch_bounds__(NTHR)
void cml2d_fused(const float* __restrict__ drive,
                 const float* __restrict__ Kw,
                 float* __restrict__ out) {
  __shared__ float sG[LDSN];   // grid state (interior 158x158 used)
  __shared__ float sM[LDSN];   // mapped, with zero guard ring (160x160)
  __shared__ float sD[LDSN];   // BETA*drive (0 outside image)

  const int tid   = threadIdx.x;
  const int bx    = blockIdx.x, by = blockIdx.y, plane = blockIdx.z;
  const int ch    = plane & 3;                  // C = 4
  const int ox    = bx * TOUT - HALO;
  const int oy    = by * TOUT - HALO;
  const bool edge = (bx == 0) | (bx == 3) | (by == 0) | (by == 3);

  // Folded 3x3 weights (uniform per workgroup -> scalar broadcast operands)
  const float ksc = (1.f - BETAC) * EPSC;       // 0.255
  float kw[9];
#pragma unroll
  for (int i = 0; i < 9; ++i) kw[i] = ksc * Kw[ch * 9 + i];
  kw[4] += (1.f - BETAC) * (1.f - EPSC);        // +0.595 center tap

  const float* dp = drive + (size_t)plane * PLANE;

  const int tx = tid & 31;             // 0..31 (lane)
  const int tg = tid >> 5;             // 0..15 (wave id, uniform per wave)
  const int r0 = tg * RPW;             // wave-owned contiguous row chunk
  const int r1 = (r0 + RPW < TIN) ? r0 + RPW : TIN;

  // ---- async-load 158x158 drive tile into sG (addresses clamped; OOB masked later)
  for (int r = r0; r < r1; ++r) {
    int gy = oy + r;
    gy = gy < 0 ? 0 : (gy > HIMG - 1 ? HIMG - 1 : gy);
    const float* grow = dp + gy * WIMG;
#pragma unroll
    for (int j = 0; j < 5; ++j) {
      int c = tx + (j << 5);
      if (c >= TIN) continue;
      int gx = ox + c;
      gx = gx < 0 ? 0 : (gx > WIMG - 1 ? WIMG - 1 : gx);
      async_ld_b32((uint32_t)(uintptr_t)&sG[r * PITCH + c], grow + gx);
    }
  }

  // Zero M while the async DMA is in flight (guard ring must stay zero forever).
  for (int i = tid; i < LDSN; i += NTHR) sM[i] = 0.f;

  wait_async_0();      // this wave's ASYNCcnt == 0
  __syncthreads();     // all waves' LDS writes visible

  // ---- D = BETA*drive, exactly 0 outside the real image (SAME zero-padding)
  for (int r = r0; r < r1; ++r) {
    bool rin = ((unsigned)(oy + r) < (unsigned)HIMG);
#pragma unroll
    for (int j = 0; j < 5; ++j) {
      int c = tx + (j << 5);
      if (c >= TIN) continue;
      bool inb = (!edge) || (rin & ((unsigned)(ox + c) < (unsigned)WIMG));
      int idx = r * PITCH + c;
      sD[idx] = inb ? BETAC * sG[idx] : 0.f;
    }
  }
  __syncthreads();

  // ---- 15 fused steps, entirely LDS-resident
  for (int s = 0; s < NSTEPS; ++s) {
    // pass A: mapped = R*g*(1-g); masked to 0 outside image (matches SAME padding)
#pragma unroll
    for (int j = 0; j < 5; ++j) {
      int c = tx + (j << 5);
      if (c >= TIN) continue;
      int idx = r0 * PITCH + c;
      for (int r = r0; r < r1; ++r) {
        float g = sG[idx];
        float t = RC * g;
        float m = __builtin_fmaf(-t, g, t);          // R*g - R*g*g
        if (edge) {
          if (!(((unsigned)(oy + r) < (unsigned)HIMG) &
                ((unsigned)(ox + c) < (unsigned)WIMG))) m = 0.f;
        }
        sM[idx + PITCH + 1] = m;                     // (r+1, c+1): interior of guarded M
        idx += PITCH;
      }
    }
    __syncthreads();

    // pass B: g = conv3x3(M, kw) + D with vertical rolling window.
    // Conv at row r reads guarded-M rows r..r+2; keep top two tap-rows in
    // registers, load only the new bottom row (3 contiguous floats) per cell.
#pragma unroll
    for (int j = 0; j < 5; ++j) {
      int c = tx + (j << 5);
      if (c >= TIN) continue;
      const float* mb = &sM[r0 * PITCH + c];
      float t0 = mb[0],         t1 = mb[1],         t2 = mb[2];
      float u0 = mb[PITCH + 0], u1 = mb[PITCH + 1], u2 = mb[PITCH + 2];
      int idx = r0 * PITCH + c;
      for (int r = r0; r < r1; ++r) {
        const float* mv = &sM[idx + 2 * PITCH];      // guarded-M row r+2
        float v0 = mv[0], v1 = mv[1], v2 = mv[2];
        float acc = sD[idx];
        acc = __builtin_fmaf(kw[0], t0, acc);
        acc = __builtin_fmaf(kw[1], t1, acc);
        acc = __builtin_fmaf(kw[2], t2, acc);
        acc = __builtin_fmaf(kw[3], u0, acc);
        acc = __builtin_fmaf(kw[4], u1, acc);
        acc = __builtin_fmaf(kw[5], u2, acc);
        acc = __builtin_fmaf(kw[6], v0, acc);
        acc = __builtin_fmaf(kw[7], v1, acc);
        acc = __builtin_fmaf(kw[8], v2, acc);
        sG[idx] = acc;
        t0 = u0; t1 = u1; t2 = u2;
        u0 = v0; u1 = v1; u2 = v2;
        idx += PITCH;
      }
    }
    __syncthreads();
  }

  // ---- clamp + store the exact 128x128 center (d >= HALO cells are bit-exact)
  float* op = out + (size_t)plane * PLANE;
  const int q  = tid & 31;   // column quad 0..31 (32*4 = 128 cols)
  const int rg = tid >> 5;   // 0..15
#pragma unroll
  for (int k = 0; k < 8; ++k) {
    int row = rg + (k << 4);             // 0..127
    const float* gr = &sG[(HALO + row) * PITCH + (HALO + (q << 2))];
    float4 v;
    v.x = fminf(fmaxf(gr[0], CLAMPC), 1.f - CLAMPC);
    v.y = fminf(fmaxf(gr[1], CLAMPC), 1.f - CLAMPC);
    v.z = fminf(fmaxf(gr[2], CLAMPC), 1.f - CLAMPC);
    v.w = fminf(fmaxf(gr[3], CLAMPC), 1.f - CLAMPC);
    *(float4*)(op + (size_t)(by * TOUT + row) * WIMG + bx * TOUT + (q << 2)) = v;
  }
}

extern "C" void kernel_launch(void* const* d_in, const int* in_sizes, int n_in,
                              void* d_out, int out_size, void* d_ws, size_t ws_size,
                              hipStream_t stream) {
  (void)in_sizes; (void)n_in; (void)d_ws; (void)ws_size; (void)out_size;
  const float* drive = (const float*)d_in[0];  // [64,4,512,512] f32
  const float* Kw    = (const float*)d_in[1];  // [4,1,3,3] f32
  float* out         = (float*)d_out;          // [64,4,512,512] f32

  dim3 grid(WIMG / TOUT, HIMG / TOUT, 64 * 4); // 4 x 4 x 256 tiles
  dim3 block(NTHR);                            // 16 wave32 waves, 1 WGP-filling block
  hipLaunchKernelGGL(cml2d_fused, grid, block, 0, stream, drive, Kw, out);
}